// MambaBlock_87179246175007
// MI455X (gfx1250) — compile-verified
//
#include <hip/hip_runtime.h>
#include <hip/hip_bf16.h>

// ---------------------------------------------------------------------------
// Mamba block forward for MI455X (gfx1250, wave32, WMMA).
// Shapes: B=2, T=1024, D_MODEL=1024, D_INNER=2048, D_STATE=16, D_CONV=4.
// ---------------------------------------------------------------------------

#define BB      2
#define TT      1024
#define DM      1024
#define DI      2048
#define DS      16
#define ROWS    (BB * TT)        // 2048 token rows

typedef float v2f __attribute__((ext_vector_type(2)));
typedef float v8f __attribute__((ext_vector_type(8)));

// ---------------------------------------------------------------------------
// 1) LayerNorm: one block per token row of 1024.
// ---------------------------------------------------------------------------
__global__ void ln_kernel(const float* __restrict__ x,
                          const float* __restrict__ gamma,
                          const float* __restrict__ beta,
                          float* __restrict__ xn) {
    const int row = blockIdx.x;
    const float* xr = x + (size_t)row * DM;

    float s = 0.f, ss = 0.f;
    for (int i = threadIdx.x; i < DM; i += 256) {
        float v = xr[i];
        s += v; ss += v * v;
    }
    // wave32 reduce
    for (int off = 16; off; off >>= 1) {
        s  += __shfl_down(s,  off, 32);
        ss += __shfl_down(ss, off, 32);
    }
    __shared__ float red0[8], red1[8];
    const int wave = threadIdx.x >> 5;
    if ((threadIdx.x & 31) == 0) { red0[wave] = s; red1[wave] = ss; }
    __syncthreads();
    if (threadIdx.x == 0) {
        float a = 0.f, b2 = 0.f;
        for (int w = 0; w < 8; ++w) { a += red0[w]; b2 += red1[w]; }
        float mu = a * (1.0f / DM);
        red0[0] = mu;
        red1[0] = b2 * (1.0f / DM) - mu * mu;
    }
    __syncthreads();
    const float mu  = red0[0];
    const float inv = rsqrtf(red1[0] + 1e-5f);
    float* xo = xn + (size_t)row * DM;
    for (int i = threadIdx.x; i < DM; i += 256)
        xo[i] = (xr[i] - mu) * inv * gamma[i] + beta[i];
}

// ---------------------------------------------------------------------------
// 2) Generic C[M,N] = A[M,K] @ W[N,K]^T (+ optional residual), fp32 WMMA.
//    Each wave computes a 16x64 strip: one A fragment feeds 4 accumulators
//    (4x V_WMMA_F32_16X16X4_F32 per K-step), cutting A traffic 4x.
//    A-frag layout (16x4 f32): lane<16 -> K={0,1}, lane>=16 -> K={2,3}.
//    B-frag layout (4x16 f32): mirrored, N indexed by lane&15.
//    Requires N % 64 == 0 (true for 4096 and 1024).
// ---------------------------------------------------------------------------
__global__ void gemm_wmma_kernel(const float* __restrict__ A,
                                 const float* __restrict__ W,
                                 const float* __restrict__ resid,
                                 float* __restrict__ C,
                                 int M, int N, int K) {
    const int wave = threadIdx.x >> 5;
    const int lane = threadIdx.x & 31;
    const int job  = blockIdx.x * (blockDim.x >> 5) + wave;
    const int grpN = N >> 6;                  // 64-wide column groups
    const int tm   = job / grpN;
    const int tg   = job - tm * grpN;
    if (tm * 16 >= M) return;                 // wave-uniform: EXEC stays all-ones

    const int l  = lane & 15;
    const int hi = lane >> 4;

    const float* arow  = A + (size_t)(tm * 16 + l) * K + 2 * hi;
    const float* brow0 = W + (size_t)(tg * 64 +  0 + l) * K + 2 * hi;
    const float* brow1 = W + (size_t)(tg * 64 + 16 + l) * K + 2 * hi;
    const float* brow2 = W + (size_t)(tg * 64 + 32 + l) * K + 2 * hi;
    const float* brow3 = W + (size_t)(tg * 64 + 48 + l) * K + 2 * hi;

    v8f acc0 = {0.f,0.f,0.f,0.f,0.f,0.f,0.f,0.f};
    v8f acc1 = acc0, acc2 = acc0, acc3 = acc0;

    for (int k = 0; k < K; k += 8) {
        if (k + 128 < K) {   // WGP-scope prefetch of upcoming K-block
            __builtin_prefetch(arow  + k + 128, 0, 3);
            __builtin_prefetch(brow0 + k + 128, 0, 3);
            __builtin_prefetch(brow1 + k + 128, 0, 3);
            __builtin_prefetch(brow2 + k + 128, 0, 3);
            __builtin_prefetch(brow3 + k + 128, 0, 3);
        }
#pragma unroll
        for (int kk = 0; kk < 8; kk += 4) {
            v2f a  = *(const v2f*)(arow  + k + kk);
            v2f b0 = *(const v2f*)(brow0 + k + kk);
            v2f b1 = *(const v2f*)(brow1 + k + kk);
            v2f b2 = *(const v2f*)(brow2 + k + kk);
            v2f b3 = *(const v2f*)(brow3 + k + kk);
            acc0 = __builtin_amdgcn_wmma_f32_16x16x4_f32(
                false, a, false, b0, (short)0, acc0, false, false);
            acc1 = __builtin_amdgcn_wmma_f32_16x16x4_f32(
                false, a, false, b1, (short)0, acc1, false, false);
            acc2 = __builtin_amdgcn_wmma_f32_16x16x4_f32(
                false, a, false, b2, (short)0, acc2, false, false);
            acc3 = __builtin_amdgcn_wmma_f32_16x16x4_f32(
                false, a, false, b3, (short)0, acc3, false, false);
        }
    }

    // C/D layout: lane<16 -> rows tm*16+v, lane>=16 -> rows tm*16+8+v, col = l.
    const int row0 = tm * 16 + hi * 8;
    const int colb = tg * 64 + l;
#pragma unroll
    for (int v = 0; v < 8; ++v) {
        const size_t r = (size_t)(row0 + v) * N;
        float o0 = acc0[v], o1 = acc1[v], o2 = acc2[v], o3 = acc3[v];
        if (resid) {
            o0 += resid[r + colb];
            o1 += resid[r + colb + 16];
            o2 += resid[r + colb + 32];
            o3 += resid[r + colb + 48];
        }
        C[r + colb]      = o0;
        C[r + colb + 16] = o1;
        C[r + colb + 32] = o2;
        C[r + colb + 48] = o3;
    }
}

// ---------------------------------------------------------------------------
// 3) Depthwise causal conv (k=4) + bias + SiLU. One thread per (b,t,d).
// ---------------------------------------------------------------------------
__global__ void conv_silu_kernel(const float* __restrict__ xz,
                                 const float* __restrict__ conv_w,
                                 const float* __restrict__ conv_b,
                                 float* __restrict__ xc) {
    const int idx = blockIdx.x * 256 + threadIdx.x;   // over ROWS*DI
    const int d  = idx & (DI - 1);
    const int bt = idx >> 11;
    const int t  = bt & (TT - 1);
    const int b  = bt >> 10;

    const float4 w = *(const float4*)(conv_w + (size_t)d * 4);
    float acc = conv_b[d];
#pragma unroll
    for (int k = 0; k < 4; ++k) {
        const int tau = t - 3 + k;
        float xv = 0.f;
        if (tau >= 0)
            xv = xz[((size_t)(b * TT + tau)) * (2 * DI) + d];  // xb = first half
        acc += xv * ((const float*)&w)[k];
    }
    const float s = acc / (1.f + __expf(-acc));       // silu
    xc[idx] = s;
}

// ---------------------------------------------------------------------------
// 4) ssm_p = xc @ W_x^T (N = 33 = 1 + 16 + 16). One block per token row,
//    xc row staged in LDS, one wave per group of output features.
// ---------------------------------------------------------------------------
__global__ void ssm_proj_kernel(const float* __restrict__ xc,
                                const float* __restrict__ Wx,
                                float* __restrict__ dtr,
                                float* __restrict__ Bp,
                                float* __restrict__ Cp) {
    __shared__ float row[DI];
    const int bt = blockIdx.x;
    const float* xr = xc + (size_t)bt * DI;
    for (int i = threadIdx.x * 4; i < DI; i += 256 * 4)
        *(float4*)&row[i] = *(const float4*)&xr[i];
    __syncthreads();

    const int wave = threadIdx.x >> 5;
    const int lane = threadIdx.x & 31;
    for (int j = wave; j < 2 * DS + 1; j += 8) {
        const float* w = Wx + (size_t)j * DI;
        float s = 0.f;
        for (int d = lane; d < DI; d += 32) s += row[d] * w[d];
        for (int off = 16; off; off >>= 1) s += __shfl_down(s, off, 32);
        if (lane == 0) {
            if (j == 0)            dtr[bt] = s;
            else if (j <= DS)      Bp[bt * DS + (j - 1)] = s;
            else                   Cp[bt * DS + (j - 1 - DS)] = s;
        }
    }
}

// ---------------------------------------------------------------------------
// 5) Sequential selective scan over T, one thread per (b,d) carrying 16
//    states in registers. `b` is derived from blockIdx so the per-timestep
//    dtr/Bp/Cp broadcasts become wave-uniform (scalar) loads.
//    Fuses +D*xc and the SiLU(z) gate, writes G.
// ---------------------------------------------------------------------------
__global__ void scan_kernel(const float* __restrict__ dtr,
                            const float* __restrict__ Bp,
                            const float* __restrict__ Cp,
                            const float* __restrict__ xc,
                            const float* __restrict__ xz,
                            const float* __restrict__ w_dt,
                            const float* __restrict__ b_dt,
                            const float* __restrict__ A_log,
                            const float* __restrict__ D_param,
                            float* __restrict__ G) {
    // 16 blocks of 256: blocks [0,8) -> b=0, [8,16) -> b=1 (uniform per block)
    const int b = blockIdx.x >> 3;
    const int d = (blockIdx.x & 7) * 256 + threadIdx.x;   // 0 .. DI-1

    const float wd = w_dt[d];
    const float bd = b_dt[d];
    const float Dp = D_param[d];

    float A[DS], h[DS];
#pragma unroll
    for (int n = 0; n < DS; ++n) {
        A[n] = -__expf(A_log[(size_t)d * DS + n]);
        h[n] = 0.f;
    }

    for (int t = 0; t < TT; ++t) {
        const int bt = b * TT + t;                         // uniform
        const float sp = dtr[bt] * wd + bd;
        const float dt = (sp > 20.f) ? sp : __logf(1.f + __expf(sp));
        const float xv = xc[(size_t)bt * DI + d];
        const float dtx = dt * xv;

        const float4* Bv4 = (const float4*)(Bp + (size_t)bt * DS);
        const float4* Cv4 = (const float4*)(Cp + (size_t)bt * DS);

        float y = 0.f;
#pragma unroll
        for (int q = 0; q < 4; ++q) {
            const float4 Bv = Bv4[q];
            const float4 Cv = Cv4[q];
            const int n = q * 4;
            h[n + 0] = __expf(dt * A[n + 0]) * h[n + 0] + dtx * Bv.x; y += h[n + 0] * Cv.x;
            h[n + 1] = __expf(dt * A[n + 1]) * h[n + 1] + dtx * Bv.y; y += h[n + 1] * Cv.y;
            h[n + 2] = __expf(dt * A[n + 2]) * h[n + 2] + dtx * Bv.z; y += h[n + 2] * Cv.z;
            h[n + 3] = __expf(dt * A[n + 3]) * h[n + 3] + dtx * Bv.w; y += h[n + 3] * Cv.w;
        }

        const float zv = xz[(size_t)bt * (2 * DI) + DI + d];   // z = second half
        const float gate = zv / (1.f + __expf(-zv));
        G[(size_t)bt * DI + d] = (y + Dp * xv) * gate;
    }
}

// ---------------------------------------------------------------------------
// Launch
// ---------------------------------------------------------------------------
extern "C" void kernel_launch(void* const* d_in, const int* in_sizes, int n_in,
                              void* d_out, int out_size, void* d_ws, size_t ws_size,
                              hipStream_t stream) {
    (void)in_sizes; (void)n_in; (void)out_size; (void)ws_size;

    const float* x        = (const float*)d_in[0];
    const float* ln_gamma = (const float*)d_in[1];
    const float* ln_beta  = (const float*)d_in[2];
    const float* W_in     = (const float*)d_in[3];
    const float* conv_w   = (const float*)d_in[4];
    const float* conv_b   = (const float*)d_in[5];
    const float* W_x      = (const float*)d_in[6];
    const float* w_dt     = (const float*)d_in[7];
    const float* b_dt     = (const float*)d_in[8];
    const float* A_log    = (const float*)d_in[9];
    const float* D_param  = (const float*)d_in[10];
    const float* W_out    = (const float*)d_in[11];
    float* out = (float*)d_out;

    float* ws  = (float*)d_ws;
    float* xn  = ws;                           // ROWS*DM      =  2,097,152
    float* xz  = xn  + (size_t)ROWS * DM;      // ROWS*2*DI    =  8,388,608
    float* xc  = xz  + (size_t)ROWS * 2 * DI;  // ROWS*DI      =  4,194,304
    float* dtr = xc  + (size_t)ROWS * DI;      // ROWS
    float* Bpv = dtr + ROWS;                   // ROWS*DS
    float* Cpv = Bpv + (size_t)ROWS * DS;      // ROWS*DS
    float* G   = Cpv + (size_t)ROWS * DS;      // ROWS*DI      =  4,194,304

    // 1) LayerNorm
    ln_kernel<<<ROWS, 256, 0, stream>>>(x, ln_gamma, ln_beta, xn);

    // 2) xz = xn @ W_in^T : M=2048, N=4096, K=1024
    //    16x64 strip per wave -> 128 * 64 = 8192 wave-jobs, 4 waves/block
    {
        const int jobs = (ROWS / 16) * ((2 * DI) / 64);
        gemm_wmma_kernel<<<jobs / 4, 128, 0, stream>>>(
            xn, W_in, nullptr, xz, ROWS, 2 * DI, DM);
    }

    // 3) depthwise conv + SiLU
    conv_silu_kernel<<<(ROWS * DI) / 256, 256, 0, stream>>>(xz, conv_w, conv_b, xc);

    // 4) ssm_p = xc @ W_x^T (split into dt_raw / B / C)
    ssm_proj_kernel<<<ROWS, 256, 0, stream>>>(xc, W_x, dtr, Bpv, Cpv);

    // 5) sequential scan + gate -> G
    scan_kernel<<<(BB * DI) / 256, 256, 0, stream>>>(
        dtr, Bpv, Cpv, xc, xz, w_dt, b_dt, A_log, D_param, G);

    // 6) out = x + G @ W_out^T : M=2048, N=1024, K=2048 -> 128*16 = 2048 jobs
    {
        const int jobs = (ROWS / 16) * (DM / 64);
        gemm_wmma_kernel<<<jobs / 4, 128, 0, stream>>>(
            G, W_out, x, out, ROWS, DM, DI);
    }
}